// MPClayer_781684048022
// MI455X (gfx1250) — compile-verified
//
#include <hip/hip_runtime.h>

typedef __attribute__((ext_vector_type(2))) float v2f;
typedef __attribute__((ext_vector_type(8))) float v8f;

namespace {
constexpr int cNH = 25;          // nHidden
constexpr int cNS = 20;          // nStep
constexpr int cD  = 27;          // augmented state dim
constexpr int cKP = 28;          // K padded to multiple of 4
constexpr int cNP = 32;          // N padded to two 16-wide WMMA tiles
constexpr float cDT  = 1.0f / 60.0f;
constexpr float cEPS = 1e-4f;
constexpr float cPQ  = 5.0f;
constexpr float cQV  = 200.0f;
constexpr float cQA  = 1.0f;
constexpr int ROWS_PER_BLOCK = 256;
constexpr int THREADS = 128;     // 4 waves (wave32)
}

// ---------------------------------------------------------------------------
// Setup kernel: one workgroup computes Mt[cNP][cKP] = padded M^T where
// out[b][s] = sum_d x0[b][d] * Mt[s][d].
// M = Tsel^T - p_final * Qinv * Ssel^T, all built from Af, Lq.
// ---------------------------------------------------------------------------
__global__ __launch_bounds__(THREADS)
void mpc_setup(const float* __restrict__ Af, const float* __restrict__ Lq,
               float* __restrict__ Mt_g) {
  __shared__ float A0[cD][cD];
  __shared__ float Pc[cD][cD];
  __shared__ float Pn[cD][cD];
  __shared__ float Q0f[cD][cD];
  __shared__ float SB[cNS + 1][cD];    // SB[k] = A0^k @ B0
  __shared__ float QSB[cNS + 1][cD];   // Q0f @ SB[k]
  __shared__ float g[cNS + 1];         // SB[k][NH]
  __shared__ float Tsel[cNS][cD];      // row NH of A0^(s+1)
  __shared__ float Hm[cNS][cNS];       // SB[a]^T Q0f SB[b]
  __shared__ float Qm[cNS][2 * cNS];   // augmented [Q_final | I] -> [I | Qinv]
  __shared__ float fac[cNS];
  __shared__ float spiv;
  __shared__ float pfin[cD][cNS];      // p_final (without final *2)
  __shared__ float Wm[cNS][cNS];       // Qinv @ Ssel^T

  const int tid = threadIdx.x;
  const float hdt2 = 0.5f * cDT * cDT;

  // ---- Q0f, A0, Pc = I, pfin = 0 ----
  for (int idx = tid; idx < cD * cD; idx += THREADS) {
    const int i = idx / cD, j = idx % cD;
    float q = 0.f;
    if (i < cNH && j < cNH) {
      float s = 0.f;
      for (int k = 0; k < cNH; ++k) s += Lq[i * cNH + k] * Lq[j * cNH + k];
      q = s + (i == j ? cEPS : 0.f);
    } else if (i == cNH + 1 && j == cNH + 1) {
      q = cQV;
    }
    Q0f[i][j] = q;

    float a = 0.f;
    if (i < cNH) {
      if (j == i) a = 1.f;
      else if (j == cD - 1) a = Af[i];
    } else if (i == cNH) {
      if (j == cNH) a = 1.f;
      else if (j == cNH + 1) a = cDT;
    } else {
      if (j == cNH + 1) a = 1.f;
    }
    A0[i][j] = a;
    Pc[i][j] = (i == j) ? 1.f : 0.f;
  }
  for (int idx = tid; idx < cD * cNS; idx += THREADS)
    (&pfin[0][0])[idx] = 0.f;
  if (tid < cD) {
    float b = 0.f;
    if (tid == cNH)     b = hdt2;
    if (tid == cNH + 1) b = cDT;
    SB[0][tid] = b;
  }
  if (tid == 0) g[0] = hdt2;
  __syncthreads();

  // ---- power iteration: Pn = A0^k; accumulate p_final on the fly ----
  for (int k = 1; k <= cNS; ++k) {
    // stage A: Pn = Pc @ A0 ; QSB[k-1] = Q0f @ SB[k-1]
    for (int idx = tid; idx < cD * cD; idx += THREADS) {
      const int i = idx / cD, j = idx % cD;
      float s = 0.f;
      for (int e = 0; e < cD; ++e) s += Pc[i][e] * A0[e][j];
      Pn[i][j] = s;
    }
    if (tid < cD) {
      float s = 0.f;
      for (int j = 0; j < cD; ++j) s += Q0f[tid][j] * SB[k - 1][j];
      QSB[k - 1][tid] = s;
    }
    __syncthreads();

    // stage B: pfin[:,j] += w_r * Pn^T @ QSB[r-j] (r = k-1); SB[k]; Tsel; Pc=Pn
    const float wk = (k == cNS) ? cPQ : 1.f;
    for (int idx = tid; idx < cD * cNS; idx += THREADS) {
      const int d = idx / cNS, j = idx % cNS;
      if (j <= k - 1) {
        const int m = k - 1 - j;
        float t = 0.f;
        for (int e = 0; e < cD; ++e) t += Pn[e][d] * QSB[m][e];
        pfin[d][j] += wk * t;
      }
    }
    for (int idx = tid; idx < cD * cD; idx += THREADS) {
      const int i = idx / cD, j = idx % cD;
      Pc[i][j] = Pn[i][j];
    }
    if (tid < cD) {
      const float sb = Pn[tid][cNH] * hdt2 + Pn[tid][cNH + 1] * cDT;
      SB[k][tid] = sb;
      Tsel[k - 1][tid] = Pn[cNH][tid];
      if (tid == cNH) g[k] = sb;
    }
    __syncthreads();
  }

  // ---- H[a][b] = SB[a] . QSB[b] ----
  for (int idx = tid; idx < cNS * cNS; idx += THREADS) {
    const int a = idx / cNS, b = idx % cNS;
    float s = 0.f;
    for (int i = 0; i < cD; ++i) s += SB[a][i] * QSB[b][i];
    Hm[a][b] = s;
  }
  __syncthreads();

  // ---- Q_final (augmented with identity) ----
  for (int idx = tid; idx < cNS * 2 * cNS; idx += THREADS) {
    const int i = idx / (2 * cNS), j = idx % (2 * cNS);
    if (j < cNS) {
      float s = 0.f;
      const int lo = i > j ? i : j;
      for (int r = lo; r < cNS; ++r) {
        const float w = (r == cNS - 1) ? cPQ : 1.f;
        s += w * Hm[r - i][r - j];
      }
      float v = 2.f * s;
      if (i == j) v += 2.f * cQA + cEPS;
      Qm[i][j] = v;
    } else {
      Qm[i][j] = ((j - cNS) == i) ? 1.f : 0.f;
    }
  }
  __syncthreads();

  // ---- Gauss-Jordan inverse (SPD, no pivoting) ----
  for (int k = 0; k < cNS; ++k) {
    if (tid == 0) spiv = 1.f / Qm[k][k];
    __syncthreads();
    if (tid < 2 * cNS) Qm[k][tid] *= spiv;
    __syncthreads();
    if (tid < cNS) fac[tid] = Qm[tid][k];
    __syncthreads();
    for (int idx = tid; idx < cNS * 2 * cNS; idx += THREADS) {
      const int i = idx / (2 * cNS), j = idx % (2 * cNS);
      if (i != k) Qm[i][j] -= fac[i] * Qm[k][j];
    }
    __syncthreads();
  }

  // ---- W[j][s] = sum_{c<=s} Qinv[j][c] * g[s-c]  (Ssel is Toeplitz) ----
  for (int idx = tid; idx < cNS * cNS; idx += THREADS) {
    const int j = idx / cNS, s = idx % cNS;
    float acc = 0.f;
    for (int c = 0; c <= s; ++c) acc += Qm[j][cNS + c] * g[s - c];
    Wm[j][s] = acc;
  }
  __syncthreads();

  // ---- Mt[s][d] = Tsel[s][d] - 2 * sum_j pfin[d][j] * W[j][s], padded ----
  for (int idx = tid; idx < cNP * cKP; idx += THREADS) {
    const int s = idx / cKP, d = idx % cKP;
    float v = 0.f;
    if (s < cNS && d < cD) {
      float acc = 0.f;
      for (int j = 0; j < cNS; ++j) acc += pfin[d][j] * Wm[j][s];
      v = Tsel[s][d] - 2.f * acc;
    }
    Mt_g[idx] = v;
  }
}

// ---------------------------------------------------------------------------
// Main kernel: out[B,20] = x0[B,27] @ M via v_wmma_f32_16x16x4_f32.
// Block: 128 threads (4 waves). Each wave: 4 row-tiles x 2 col-tiles.
// ---------------------------------------------------------------------------
__global__ __launch_bounds__(THREADS)
void mpc_main(const float* __restrict__ x, const float* __restrict__ gp,
              const float* __restrict__ gv, const float* __restrict__ Mt_g,
              float* __restrict__ out) {
  __shared__ float xs[ROWS_PER_BLOCK][cKP];   // 28 KB, bank-conflict-free stride
  __shared__ float Ms[cNP][cKP];              // M^T padded, 3.5 KB

  const int tid = threadIdx.x;
  const int b0 = blockIdx.x * ROWS_PER_BLOCK;

  for (int idx = tid; idx < cNP * cKP; idx += THREADS)
    (&Ms[0][0])[idx] = Mt_g[idx];
  for (int idx = tid; idx < ROWS_PER_BLOCK * cNH; idx += THREADS) {
    const int r = idx / cNH, c = idx % cNH;
    xs[r][c] = x[(size_t)(b0 + r) * cNH + c];
  }
  for (int r = tid; r < ROWS_PER_BLOCK; r += THREADS) {
    xs[r][cNH]     = gp[b0 + r];
    xs[r][cNH + 1] = gv[b0 + r];
    xs[r][cNH + 2] = 0.f;
  }
  __syncthreads();

  const int wave = tid >> 5;       // wave32
  const int lane = tid & 31;
  const int lr = lane & 15;
  const int hi = lane >> 4;        // half-wave: K pair select
  const int wr0 = wave * 64;       // 64 batch rows per wave

  v8f acc[4][2];
  const v8f vzero = {0.f, 0.f, 0.f, 0.f, 0.f, 0.f, 0.f, 0.f};
#pragma unroll
  for (int rt = 0; rt < 4; ++rt) {
    acc[rt][0] = vzero;
    acc[rt][1] = vzero;
  }

#pragma unroll
  for (int kk = 0; kk < cKP / 4; ++kk) {
    const int k0 = kk * 4 + 2 * hi;
    // B fragments: lane holds (B[k0][n], B[k0+1][n]) for its column n
    const v2f bf0 = *(const v2f*)&Ms[lr][k0];
    const v2f bf1 = *(const v2f*)&Ms[lr + 16][k0];
#pragma unroll
    for (int rt = 0; rt < 4; ++rt) {
      const v2f af = *(const v2f*)&xs[wr0 + rt * 16 + lr][k0];
      acc[rt][0] = __builtin_amdgcn_wmma_f32_16x16x4_f32(
          false, af, false, bf0, (short)0, acc[rt][0], false, false);
      acc[rt][1] = __builtin_amdgcn_wmma_f32_16x16x4_f32(
          false, af, false, bf1, (short)0, acc[rt][1], false, false);
    }
  }

  // C/D layout: VGPR r -> row (r + 8*hi) of tile, lanes 0..15 -> columns
#pragma unroll
  for (int rt = 0; rt < 4; ++rt) {
    const int rowbase = b0 + wr0 + rt * 16 + hi * 8;
#pragma unroll
    for (int r = 0; r < 8; ++r)
      out[(size_t)(rowbase + r) * cNS + lr] = acc[rt][0][r];
    if (lr < cNS - 16) {
#pragma unroll
      for (int r = 0; r < 8; ++r)
        out[(size_t)(rowbase + r) * cNS + 16 + lr] = acc[rt][1][r];
    }
  }
}

// ---------------------------------------------------------------------------
extern "C" void kernel_launch(void* const* d_in, const int* in_sizes, int n_in,
                              void* d_out, int out_size, void* d_ws, size_t ws_size,
                              hipStream_t stream) {
  const float* x  = (const float*)d_in[0];
  const float* gp = (const float*)d_in[1];
  const float* gv = (const float*)d_in[2];
  const float* Af = (const float*)d_in[3];
  const float* Lq = (const float*)d_in[4];
  float* Mt = (float*)d_ws;                       // cNP*cKP floats

  mpc_setup<<<1, THREADS, 0, stream>>>(Af, Lq, Mt);

  const int B = in_sizes[1];                      // gripper_p count == BATCH
  const int blocks = B / ROWS_PER_BLOCK;          // 131072/256 = 512
  mpc_main<<<blocks, THREADS, 0, stream>>>(x, gp, gv, Mt, (float*)d_out);
}